// TCVAE_87411174408435
// MI455X (gfx1250) — compile-verified
//
#include <hip/hip_runtime.h>

#define LOG2PI 1.8378770664093453f

typedef __attribute__((ext_vector_type(2))) float v2f;
typedef __attribute__((ext_vector_type(8))) float v8f;

// ---------------------------------------------------------------------------
// Kernel 1: partial sums of |x - recon|
// ---------------------------------------------------------------------------
__global__ void recon_partial_kernel(const float* __restrict__ x,
                                     const float* __restrict__ r,
                                     float* __restrict__ part, int n) {
  __shared__ float sm[256];
  float acc = 0.0f;
  for (int i = blockIdx.x * blockDim.x + threadIdx.x; i < n;
       i += gridDim.x * blockDim.x)
    acc += fabsf(x[i] - r[i]);
  sm[threadIdx.x] = acc;
  __syncthreads();
  for (int o = 128; o > 0; o >>= 1) {
    if ((int)threadIdx.x < o) sm[threadIdx.x] += sm[threadIdx.x + o];
    __syncthreads();
  }
  if (threadIdx.x == 0) part[blockIdx.x] = sm[0];
}

// ---------------------------------------------------------------------------
// Kernel 2: per-row prep.  One thread per row j (D = 16 fixed).
//   Amat[j][0:16]  = z^2            Amat[j][16:32] = z
//   Bmat[j][0:16]  = exp(-lv)       Bmat[j][16:32] = -2*mu*exp(-lv)
//   Pack[j][0:16]  = mu   [16:32] = lv   [32:48] = exp(-lv)
//   c[j] = sum_d (mu^2*inv + lv)
//   lqzx[j] = log q(z_j|x_j),  lprior[j] = log p(z_j)  (ones-as-logvar quirk)
// ---------------------------------------------------------------------------
__global__ void prep_kernel(const float* __restrict__ mu,
                            const float* __restrict__ lv,
                            const float* __restrict__ z,
                            float* __restrict__ Amat, float* __restrict__ Bmat,
                            float* __restrict__ Pack, float* __restrict__ cvec,
                            float* __restrict__ lqzx,
                            float* __restrict__ lprior, int B) {
  int j = blockIdx.x * blockDim.x + threadIdx.x;
  if (j >= B) return;
  const float INVE = 0.36787944117144233f;  // e^-1
  float c = 0.0f, q = 0.0f, p = 0.0f;
#pragma unroll
  for (int d = 0; d < 16; ++d) {
    float m = mu[j * 16 + d];
    float l = lv[j * 16 + d];
    float zz = z[j * 16 + d];
    float inv = __expf(-l);
    Bmat[j * 32 + d] = inv;
    Bmat[j * 32 + 16 + d] = -2.0f * m * inv;
    Amat[j * 32 + d] = zz * zz;
    Amat[j * 32 + 16 + d] = zz;
    Pack[j * 48 + d] = m;
    Pack[j * 48 + 16 + d] = l;
    Pack[j * 48 + 32 + d] = inv;
    c += m * m * inv + l;
    float dz = zz - m;
    q += -0.5f * (dz * dz * inv + l + LOG2PI);
    p += -0.5f * (zz * zz * INVE + 1.0f + LOG2PI);
  }
  cvec[j] = c;
  lqzx[j] = q;
  lprior[j] = p;
}

// ---------------------------------------------------------------------------
// Kernel 3: fused WMMA GEMM + row-wise logsumexp -> log_qz[i]
// 4 waves per block; each wave handles every 4th 16-col tile of a 16-row
// strip.  Tile 0 is peeled (importance-weight special columns); the steady
// loop uses a single constant weight and a branchless online LSE, so EXEC
// stays all-1s around every v_wmma.
//
// Assumed f32 WMMA lane layouts (CDNA5 ISA 7.12.2):
//   A (16x4):  VGPR v, lanes 0-15 -> (M=lane, K=v);    lanes 16-31 -> (M, K=v+2)
//   B (4x16):  VGPR v, lanes 0-15 -> (K=v, N=lane);    lanes 16-31 -> (K=v+2, N)
//   C (16x16): VGPR v, lanes 0-15 -> (M=v, N=lane);    lanes 16-31 -> (M=v+8, N)
// ---------------------------------------------------------------------------
__global__ void wmma_logqz_kernel(const float* __restrict__ Amat,
                                  const float* __restrict__ Bmat,
                                  const float* __restrict__ cvec,
                                  const int* __restrict__ sizeN,
                                  float* __restrict__ logqz, int B) {
  const int tid = threadIdx.x;
  const int lane = tid & 31;
  const int w = tid >> 5;  // wave id 0..3
  const int l16 = lane & 15;
  const int half = lane >> 4;
  const int i_base = blockIdx.x * 16;

  // importance-weight constants
  const float Nf = (float)(*sizeN);
  const float Mf = (float)(B - 1);
  const float strat = (Nf - Mf) / (Nf * Mf);
  const float lw0 = -__logf(Nf);
  const float lws = __logf(strat);
  const float lwd = -__logf(Mf);

  // load A strip into WMMA register layout (8 chunks of K=4 -> 16 floats)
  float a[16];
  const int arow = i_base + l16;
#pragma unroll
  for (int kc = 0; kc < 8; ++kc) {
    a[2 * kc + 0] = Amat[arow * 32 + 4 * kc + 2 * half + 0];
    a[2 * kc + 1] = Amat[arow * 32 + 4 * kc + 2 * half + 1];
  }

  float m[8], s[8];
#pragma unroll
  for (int v = 0; v < 8; ++v) { m[v] = -3.0e38f; s[v] = 0.0f; }

  auto tile = [&](int jt, bool first) {
    const int jb = jt * 16;
    const int jcol = jb + l16;

    float b[16];
#pragma unroll
    for (int kc = 0; kc < 8; ++kc) {
      b[2 * kc + 0] = Bmat[jcol * 32 + 4 * kc + 2 * half + 0];
      b[2 * kc + 1] = Bmat[jcol * 32 + 4 * kc + 2 * half + 1];
    }

    v8f acc = {0.f, 0.f, 0.f, 0.f, 0.f, 0.f, 0.f, 0.f};
#pragma unroll
    for (int kc = 0; kc < 8; ++kc) {
      v2f av = {a[2 * kc + 0], a[2 * kc + 1]};
      v2f bv = {b[2 * kc + 0], b[2 * kc + 1]};
      acc = __builtin_amdgcn_wmma_f32_16x16x4_f32(
          false, av, false, bv, (short)0, acc, false, false);
    }

    const float base = -0.5f * cvec[jcol] - 8.0f * LOG2PI;
    float lwcol = lwd;
    if (first) lwcol = (jcol >= 2) ? lwd : ((jcol == 1) ? lws : lw0);

#pragma unroll
    for (int v = 0; v < 8; ++v) {
      float lw = lwcol;
      if (first) {
        const int i = i_base + v + 8 * half;
        if (jcol == 0 && i == B - 2) lw = lws;
      }
      const float x = -0.5f * acc[v] + base + lw;
      // branchless online logsumexp
      const float mn = fmaxf(m[v], x);
      s[v] = s[v] * __expf(m[v] - mn) + __expf(x - mn);
      m[v] = mn;
    }
  };

  const int ntiles = B >> 4;
  if (w == 0) tile(0, true);
  for (int jt = (w == 0 ? 4 : w); jt < ntiles; jt += 4) tile(jt, false);

  // combine (m,s) across the 16 lanes holding the same rows
#pragma unroll
  for (int v = 0; v < 8; ++v) {
    for (int off = 1; off < 16; off <<= 1) {
      float mo = __shfl_xor(m[v], off, 16);
      float so = __shfl_xor(s[v], off, 16);
      float mn = fmaxf(m[v], mo);
      s[v] = s[v] * __expf(m[v] - mn) + so * __expf(mo - mn);
      m[v] = mn;
    }
  }

  // combine across the 4 waves via LDS
  __shared__ float smM[4][16];
  __shared__ float smS[4][16];
  if (l16 == 0) {
#pragma unroll
    for (int v = 0; v < 8; ++v) {
      const int row = v + 8 * half;
      smM[w][row] = m[v];
      smS[w][row] = s[v];
    }
  }
  __syncthreads();
  if (tid < 16) {
    float mm = smM[0][tid], ss = smS[0][tid];
#pragma unroll
    for (int q = 1; q < 4; ++q) {
      const float mo = smM[q][tid], so = smS[q][tid];
      const float mn = fmaxf(mm, mo);
      ss = ss * __expf(mm - mn) + so * __expf(mo - mn);
      mm = mn;
    }
    logqz[i_base + tid] = mm + __logf(ss);
  }
}

// ---------------------------------------------------------------------------
// Kernel 4: dimension-wise logsumexp.  One thread per (i,d); loops over j.
// j = 0,1 peeled (special importance weights); steady loop reads mu/lv/inv
// from an LDS-staged packed array and uses a branchless online LSE.
// Block = 256 threads = 16 consecutive i * 16 d; they share the j stream.
// ---------------------------------------------------------------------------
#define JC 64  // j rows staged per LDS chunk
__global__ void dimwise_lse_kernel(const float* __restrict__ Pack,
                                   const float* __restrict__ z,
                                   const int* __restrict__ sizeN,
                                   float* __restrict__ lse, int B) {
  __shared__ float smP[JC * 48];
  const int tid = threadIdx.x;
  const int t = blockIdx.x * 256 + tid;
  const int d = t & 15;
  const int i = t >> 4;

  const float Nf = (float)(*sizeN);
  const float Mf = (float)(B - 1);
  const float strat = (Nf - Mf) / (Nf * Mf);
  const float lw0 = -__logf(Nf);
  const float lws = __logf(strat);
  const float lwd = -__logf(Mf);

  const float zi = z[i * 16 + d];
  float m = -3.0e38f, s = 0.0f;

  // prologue: j = 0 and j = 1 with special weights (direct global reads)
#pragma unroll
  for (int j = 0; j < 2; ++j) {
    const float mj = Pack[j * 48 + d];
    const float lj = Pack[j * 48 + 16 + d];
    const float inv = Pack[j * 48 + 32 + d];
    const float lw = (j == 1) ? lws : ((i == B - 2) ? lws : lw0);
    const float dz = zi - mj;
    const float x = -0.5f * (dz * dz * inv + lj + LOG2PI) + lw;
    const float mn = fmaxf(m, x);
    s = s * __expf(m - mn) + __expf(x - mn);
    m = mn;
  }

  // steady loop: constant weight lwd, LDS-staged chunks
  const int nchunk = B / JC;
  for (int c = 0; c < nchunk; ++c) {
    for (int k = tid; k < JC * 48; k += 256) smP[k] = Pack[c * JC * 48 + k];
    __syncthreads();
    const int jj0 = (c == 0) ? 2 : 0;
    for (int jj = jj0; jj < JC; ++jj) {
      const float mj = smP[jj * 48 + d];
      const float lj = smP[jj * 48 + 16 + d];
      const float inv = smP[jj * 48 + 32 + d];
      const float dz = zi - mj;
      const float x = -0.5f * (dz * dz * inv + lj + LOG2PI) + lwd;
      const float mn = fmaxf(m, x);
      s = s * __expf(m - mn) + __expf(x - mn);
      m = mn;
    }
    __syncthreads();
  }
  lse[i * 16 + d] = m + __logf(s);
}

// ---------------------------------------------------------------------------
// Kernel 5: final reductions + loss assembly (single block).
// ---------------------------------------------------------------------------
__global__ void finalize_kernel(const float* __restrict__ part, int nPart,
                                const float* __restrict__ lqzx,
                                const float* __restrict__ logqz,
                                const float* __restrict__ lprior,
                                const float* __restrict__ lse, int B, int F,
                                float* __restrict__ out) {
  __shared__ float sm[256];
  const int tid = threadIdx.x;

  float acc[5] = {0.f, 0.f, 0.f, 0.f, 0.f};
  for (int i = tid; i < nPart; i += 256) acc[0] += part[i];
  for (int i = tid; i < B; i += 256) {
    acc[1] += lqzx[i];
    acc[2] += logqz[i];
    acc[3] += lprior[i];
  }
  for (int i = tid; i < B * 16; i += 256) acc[4] += lse[i];

  float tot[5];
#pragma unroll
  for (int k = 0; k < 5; ++k) {
    sm[tid] = acc[k];
    __syncthreads();
    for (int o = 128; o > 0; o >>= 1) {
      if (tid < o) sm[tid] += sm[tid + o];
      __syncthreads();
    }
    tot[k] = sm[0];
    __syncthreads();
  }

  if (tid == 0) {
    const float invB = 1.0f / (float)B;
    const float recon = tot[0] / ((float)B * (float)F);
    const float mean_q = tot[1] * invB;
    const float mean_z = tot[2] * invB;
    const float mean_p = tot[3] * invB;
    const float mean_prod = tot[4] * invB;  // lse already summed over d
    // alpha*mi + beta*tc + gamma*dwkl with alpha=gamma=1, beta=4
    const float total = recon + (mean_q - mean_z) + 4.0f * (mean_z - mean_prod)
                        + (mean_prod - mean_p);
    out[0] = total;
  }
}

// ---------------------------------------------------------------------------
extern "C" void kernel_launch(void* const* d_in, const int* in_sizes, int n_in,
                              void* d_out, int out_size, void* d_ws,
                              size_t ws_size, hipStream_t stream) {
  const float* x = (const float*)d_in[0];
  const float* recon = (const float*)d_in[1];
  const float* mu = (const float*)d_in[2];
  const float* lv = (const float*)d_in[3];
  const float* z = (const float*)d_in[4];
  const int* sizeN = (const int*)d_in[5];

  const int D = 16;
  const int B = in_sizes[2] / D;   // mu is [B, D]
  const int F = in_sizes[0] / B;   // x is [B, F]
  const int nRecon = in_sizes[0];
  const int nPart = 512;

  float* ws = (float*)d_ws;
  float* w_part = ws;                       // 512
  float* w_c = w_part + nPart;              // B
  float* w_lqzx = w_c + B;                  // B
  float* w_lprior = w_lqzx + B;             // B
  float* w_logqz = w_lprior + B;            // B
  float* w_lse = w_logqz + B;               // B*16
  float* w_A = w_lse + B * 16;              // B*32
  float* w_B = w_A + B * 32;                // B*32
  float* w_pack = w_B + B * 32;             // B*48

  recon_partial_kernel<<<nPart, 256, 0, stream>>>(x, recon, w_part, nRecon);

  prep_kernel<<<(B + 255) / 256, 256, 0, stream>>>(mu, lv, z, w_A, w_B, w_pack,
                                                   w_c, w_lqzx, w_lprior, B);

  wmma_logqz_kernel<<<B / 16, 128, 0, stream>>>(w_A, w_B, w_c, sizeN, w_logqz,
                                                B);

  dimwise_lse_kernel<<<(B * 16) / 256, 256, 0, stream>>>(w_pack, z, sizeN,
                                                         w_lse, B);

  finalize_kernel<<<1, 256, 0, stream>>>(w_part, nPart, w_lqzx, w_logqz,
                                         w_lprior, w_lse, B, F, (float*)d_out);
}